// Surface_59322088292653
// MI455X (gfx1250) — compile-verified
//
#include <hip/hip_runtime.h>
#include <math.h>

// Problem geometry (fixed by the reference): x,y are (8,1,256,256)
#define NB   8
#define NH   256
#define NW   256
#define NTOT (NB * NH * NW)   // 524288
#define FINF 1.0e12f

typedef __attribute__((ext_vector_type(2))) float v2f;
typedef __attribute__((ext_vector_type(8))) float v8f;
typedef __attribute__((ext_vector_type(4))) int   v4i;

// Async global->LDS path (gfx1250 GLOBAL_LOAD_ASYNC_TO_LDS_B128, ASYNCcnt).
#if defined(__gfx1250__) && defined(__has_builtin)
#if __has_builtin(__builtin_amdgcn_global_load_async_to_lds_b128) && \
    __has_builtin(__builtin_amdgcn_s_wait_asynccnt)
#define USE_ASYNC_LDS 1
#endif
#endif

#ifdef USE_ASYNC_LDS
// Builtin signature (from hipcc diagnostic): arg0 = AS(1) v4i* (global src),
// arg1 = AS(3) v4i* (LDS dst), then imm offset, imm cpol.
typedef __attribute__((address_space(1))) v4i* g4p_t;
typedef __attribute__((address_space(3))) v4i* l4p_t;
__device__ __forceinline__ void async_b128(const float* g, float* l) {
  __builtin_amdgcn_global_load_async_to_lds_b128(
      (g4p_t)(g), (l4p_t)(l), /*offset=*/0, /*cpol=*/0);
}
#endif

// ---------------------------------------------------------------------------
// Exact f32 wave(32) reduction using V_WMMA_F32_16X16X4_F32 with B = ones.
// A (16x4 f32) layout: lanes 0-15 {VGPR0=K0, VGPR1=K1}, lanes 16-31 {K2,K3}.
// Put the per-lane partial in a.x, zero a.y -> row m of A sums to p[m]+p[m+16].
// D = A x ones + 0 -> every column holds the 16 row sums; summing the 8 C/D
// VGPRs per lane gives rows 0-7 (lanes<16) or rows 8-15 (lanes>=16); one
// shfl_xor(16) combines the halves. Exact f32 arithmetic throughout.
// ---------------------------------------------------------------------------
__device__ __forceinline__ float wave_reduce_sum(float p) {
#if defined(__gfx1250__) && __has_builtin(__builtin_amdgcn_wmma_f32_16x16x4_f32)
  v2f a; a.x = p;    a.y = 0.0f;
  v2f b; b.x = 1.0f; b.y = 1.0f;          // all-ones B regardless of layout
  v8f c = {};
  c = __builtin_amdgcn_wmma_f32_16x16x4_f32(
      /*neg_a=*/false, a, /*neg_b=*/false, b,
      /*c_mod=*/(short)0, c, /*reuse_a=*/false, /*reuse_b=*/false);
  float s = c[0] + c[1] + c[2] + c[3] + c[4] + c[5] + c[6] + c[7];
  s += __shfl_xor(s, 16, 32);             // combine the two half-row sums
  return s;                               // every lane holds the wave total
#else
  for (int off = 16; off > 0; off >>= 1) p += __shfl_xor(p, off, 32);
  return p;
#endif
}

__device__ __forceinline__ void agent_atomic_add(float* p, float v) {
  __hip_atomic_fetch_add(p, v, __ATOMIC_RELAXED, __HIP_MEMORY_SCOPE_AGENT);
}

// ---------------------------------------------------------------------------
// Pass 0: f = (y>0) ? 0 : INF ; zero the 4 global accumulators.
// ---------------------------------------------------------------------------
__global__ void k_init(const int* __restrict__ y, float* __restrict__ f,
                       float* __restrict__ accum, int n) {
  int i = blockIdx.x * blockDim.x + threadIdx.x;
  if (i < n) f[i] = (y[i] > 0) ? 0.0f : FINF;
  if (i < 4) accum[i] = 0.0f;
}

// ---------------------------------------------------------------------------
// Pass 1: 1-D envelope along w (innermost). One block per row; row staged in
// LDS via async B128 copies (threads 0-63 = waves 0-1 fully active -> EXEC
// uniform per wave). D[i] = min_j f[j] + (i-j)^2; LDS reads broadcast.
// ---------------------------------------------------------------------------
__global__ void k_edt_row(float* __restrict__ f) {
  __shared__ float sf[NW];
  float* base = f + (size_t)blockIdx.x * NW;   // 2048 rows (b*256 + h)
  const int i = threadIdx.x;
#ifdef USE_ASYNC_LDS
  if (i < NW / 4) async_b128(base + i * 4, sf + i * 4);
  __builtin_amdgcn_s_wait_asynccnt(0);
#else
  sf[i] = base[i];
#endif
  __syncthreads();
  float m = 3.0e12f;
#pragma unroll 8
  for (int j = 0; j < NW; ++j) {
    float dj = (float)(i - j);
    m = fminf(m, sf[j] + dj * dj);
  }
  base[i] = m;
}

// ---------------------------------------------------------------------------
// Pass 2: 1-D envelope along h. Block handles a 256(h) x 32(w) tile of one
// batch image, staged in LDS (32 KB) with async B128 copies (8 per thread,
// coalesced 128B per 8 threads). Scans hit 32 distinct banks (stride-32
// tile, lane = w) -> conflict-free.
// ---------------------------------------------------------------------------
#define TW 32
__global__ void k_edt_col(float* __restrict__ f) {
  __shared__ float tile[NH * TW];
  const int b  = blockIdx.x;           // 0..7
  const int w0 = blockIdx.y * TW;      // 0..224 step 32
  float* base = f + (size_t)b * NH * NW + w0;
  const int t = threadIdx.x;
#ifdef USE_ASYNC_LDS
  {
    const int c  = t % 8;              // 16B chunk within a 32-float tile row
    const int h0 = t / 8;              // 0..31
#pragma unroll
    for (int k = 0; k < 8; ++k) {
      int h = h0 + 32 * k;
      async_b128(base + (size_t)h * NW + c * 4, tile + h * TW + c * 4);
    }
    __builtin_amdgcn_s_wait_asynccnt(0);
  }
#else
  {
    const int lw  = t % TW;
    const int lh0 = t / TW;
#pragma unroll 4
    for (int k = 0; k < NH / 8; ++k) {
      int h = lh0 + 8 * k;
      tile[h * TW + lw] = base[(size_t)h * NW + lw];
    }
  }
#endif
  __syncthreads();
  const int lw  = t % TW;              // 0..31
  const int lh0 = t / TW;              // 0..7
  for (int k = 0; k < NH / 8; ++k) {
    int i = lh0 + 8 * k;
    float m = 3.0e12f;
#pragma unroll 8
    for (int j = 0; j < NH; ++j) {
      float dj = (float)(i - j);
      m = fminf(m, tile[j * TW + lw] + dj * dj);
    }
    base[(size_t)i * NW + lw] = m;
  }
}

// ---------------------------------------------------------------------------
// Pass 3: 1-D envelope along batch (length 8, stride 65536). Fiber lives in
// registers; loads/stores coalesced across threads for each b.
// ---------------------------------------------------------------------------
__global__ void k_edt_batch(float* __restrict__ f) {
  int idx = blockIdx.x * blockDim.x + threadIdx.x;   // 0..65535
  float v[NB], o[NB];
#pragma unroll
  for (int b = 0; b < NB; ++b) v[b] = f[idx + b * (NH * NW)];
#pragma unroll
  for (int i = 0; i < NB; ++i) {
    float m = 3.0e12f;
#pragma unroll
    for (int j = 0; j < NB; ++j) {
      float d = (float)(i - j);
      m = fminf(m, v[j] + d * d);
    }
    o[i] = m;
  }
#pragma unroll
  for (int b = 0; b < NB; ++b) f[idx + b * (NH * NW)] = o[b];
}

// ---------------------------------------------------------------------------
// Pass 4: fused sums. s0=Σxs*y, s1=Σxs, s2=Σy, s3=Σxs*sqrt(f).
// Full blocks + reduction outside any divergent region -> EXEC all-ones at
// the WMMA wave reductions. One agent-scope f32 atomic per wave per sum.
// ---------------------------------------------------------------------------
__global__ void k_reduce(const float* __restrict__ x, const int* __restrict__ y,
                         const float* __restrict__ f, float* __restrict__ accum,
                         int n) {
  int tid    = blockIdx.x * blockDim.x + threadIdx.x;
  int stride = gridDim.x * blockDim.x;
  float s0 = 0.0f, s1 = 0.0f, s2 = 0.0f, s3 = 0.0f;
  for (int i = tid; i < n; i += stride) {
    if (i + stride < n) {                      // global_prefetch_b8 next tile
      __builtin_prefetch(&x[i + stride], 0, 3);
      __builtin_prefetch(&f[i + stride], 0, 3);
    }
    float xs   = 1.0f / (1.0f + expf(-x[i]));
    float yf   = (float)y[i];
    float dist = sqrtf(f[i]);
    s0 += xs * yf;
    s1 += xs;
    s2 += yf;
    s3 += xs * dist;
  }
  s0 = wave_reduce_sum(s0);
  s1 = wave_reduce_sum(s1);
  s2 = wave_reduce_sum(s2);
  s3 = wave_reduce_sum(s3);
  if ((threadIdx.x & 31) == 0) {
    agent_atomic_add(&accum[0], s0);
    agent_atomic_add(&accum[1], s1);
    agent_atomic_add(&accum[2], s2);
    agent_atomic_add(&accum[3], s3);
  }
}

// ---------------------------------------------------------------------------
// Pass 5: scalar combine.
// ---------------------------------------------------------------------------
__global__ void k_finalize(const float* __restrict__ accum,
                           float* __restrict__ out, float n) {
  float s0 = accum[0], s1 = accum[1], s2 = accum[2], s3 = accum[3];
  float dce      = 1.0f - (2.0f * s0 + 1e-5f) / (s1 + s2 + 1e-5f);
  float boundary = s3 / n;
  out[0] = dce + 0.5f * boundary;
}

extern "C" void kernel_launch(void* const* d_in, const int* in_sizes, int n_in,
                              void* d_out, int out_size, void* d_ws, size_t ws_size,
                              hipStream_t stream) {
  (void)n_in; (void)out_size; (void)ws_size;
  const float* x = (const float*)d_in[0];
  const int*   y = (const int*)d_in[1];
  const int n = in_sizes[0];               // 524288

  float* f     = (float*)d_ws;             // 2 MB distance plane
  float* accum = f + NTOT;                 // 4 global accumulators
  float* out   = (float*)d_out;

  k_init     <<<(n + 255) / 256, 256, 0, stream>>>(y, f, accum, n);
  k_edt_row  <<<NB * NH,          256, 0, stream>>>(f);          // along w
  dim3 gcol(NB, NW / TW);
  k_edt_col  <<<gcol,             256, 0, stream>>>(f);          // along h
  k_edt_batch<<<(NH * NW) / 256,  256, 0, stream>>>(f);          // along batch
  k_reduce   <<<512,              256, 0, stream>>>(x, y, f, accum, n);
  k_finalize <<<1,                1,   0, stream>>>(accum, out, (float)n);
}